// MemristorConv1d_67585605370374
// MI455X (gfx1250) — compile-verified
//
#include <hip/hip_runtime.h>
#include <stdint.h>

// ---------------- problem constants ----------------
#define Bn 4
#define Cn 512
#define Tn 400
#define Kn 31
#define TPAD (Tn + Kn - 1)        // 430 padded time length
#define TILE_T 256                // t-tile per block (one thread per t)
#define QROW (TILE_T + 32)        // 288 staged window (>= TILE_T+K-1, mult of 4)
#define NCOMBO 4                  // (i,s) pairs: (0,0),(0,1),(1,0),(1,1)
#define LUT_N 257                 // DAC levels q in [-128,128]

#define WSCALE     (0.6f / 128.0f)      // VMAX * DAC_SCALE
#define CUR_SCALEF 8020.0f
#define ADC_LIM    131072.0f            // 2^(8+10-1)
#define J_COEF     1.65677880e-28f      // 4*kB*300*BW
#define S_COEF     5.43656366e-08f      // 2*e*BW
#define EPSC       1e-12f
#define TWO_PI     6.28318530718f

// ---------------- fast TRANS ops (single v_rcp_f32 / v_sqrt_f32) ----------------
#if defined(__has_builtin)
#  if __has_builtin(__builtin_amdgcn_rcpf)
#    define FAST_RCP(x) __builtin_amdgcn_rcpf(x)
#  endif
#  if __has_builtin(__builtin_amdgcn_sqrtf)
#    define FAST_SQRT(x) __builtin_amdgcn_sqrtf(x)
#  endif
#endif
#ifndef FAST_RCP
#  define FAST_RCP(x) __frcp_rn(x)
#endif
#ifndef FAST_SQRT
#  define FAST_SQRT(x) __fsqrt_rn(x)
#endif

// ---------------- TDM availability ----------------
#if defined(__has_builtin)
#  if __has_builtin(__builtin_amdgcn_tensor_load_to_lds) && __has_builtin(__builtin_amdgcn_s_wait_tensorcnt)
#    define HAVE_TDM 1
#  endif
#endif
#ifndef HAVE_TDM
#  define HAVE_TDM 0
#endif

#if HAVE_TDM
typedef unsigned int u32x4 __attribute__((ext_vector_type(4)));
typedef int          i32x8 __attribute__((ext_vector_type(8)));
typedef int          i32x4 __attribute__((ext_vector_type(4)));

// Build a D# (groups 0/1 per CDNA5 ISA) and issue
// tensor_load_to_lds (6-arg clang-23 form; trailing group zero-filled).
// 1D when tile_d1==0, else 2D with row stride. All args are block-uniform
// -> descriptor lives in SGPRs.
__device__ __forceinline__ void tdm_load(unsigned lds_addr, const void* gptr,
                                         unsigned ds_log2,      // 0:1B 1:2B 2:4B
                                         unsigned tensor_d0,    // OOB limit, elems
                                         unsigned tile_d0,      // elems
                                         unsigned tile_d1,      // rows (0 = 1D)
                                         unsigned long long row_stride) // elems
{
  unsigned long long ga = (unsigned long long)(uintptr_t)gptr;
  unsigned td1 = tile_d1 ? tile_d1 : 1u;
  u32x4 g0;
  g0[0] = 1u;                                            // count=1 (valid), user mode
  g0[1] = lds_addr;                                      // lds_addr [63:32]
  g0[2] = (unsigned)ga;                                  // global_addr[31:0]
  g0[3] = (unsigned)((ga >> 32) & 0x01ffffffu) | (2u << 30); // addr[56:32] | type=2
  i32x8 g1;
  g1[0] = (int)(ds_log2 << 16);                          // wg_mask=0 | data_size
  g1[1] = (int)((tensor_d0 & 0xffffu) << 16);            // tensor_dim0[15:0]
  g1[2] = (int)(((tensor_d0 >> 16) & 0xffffu) | ((td1 & 0xffffu) << 16)); // d0 hi | tensor_dim1 lo
  g1[3] = (int)((tile_d0 & 0xffffu) << 16);              // tensor_dim1 hi=0 | tile_dim0
  g1[4] = (int)(tile_d1 & 0xffffu);                      // tile_dim1 | tile_dim2=0
  g1[5] = (int)(row_stride & 0xffffffffu);               // tensor_dim0_stride[31:0]
  g1[6] = (int)((row_stride >> 32) & 0xffffu);           // stride[47:32] | dim1_stride=0
  g1[7] = 0;
  i32x4 gz4 = {0, 0, 0, 0};
  i32x8 gz8 = {0, 0, 0, 0, 0, 0, 0, 0};
  __builtin_amdgcn_tensor_load_to_lds(g0, g1, gz4, gz4, gz8, 0);
}
#endif

// ---------------- RNG helpers (counter hash + Box-Muller) ----------------
__device__ __forceinline__ unsigned hash32(unsigned x) {
  x ^= x >> 16; x *= 0x7feb352dU;
  x ^= x >> 15; x *= 0x846ca68bU;
  x ^= x >> 16;
  return x;
}
__device__ __forceinline__ float u01(unsigned h) {
  return (float)(h >> 8) * 5.96046448e-08f + 2.98023224e-08f; // (0,1)
}

__device__ __forceinline__ float adc_fq(float y) {
  float q = rintf(y * 256.0f);
  q = fminf(fmaxf(q, -ADC_LIM), ADC_LIM);
  return q * (1.0f / 256.0f);
}

// ---------------- kernel A: per-level LUT (257 levels x 4 combos) ----------------
// lut_g[qi*8 + j*2 + {0,1}] = { low_j(w(qi)), high_j(w(qi)) - low_j(w(qi)) }
__global__ void mc_build_lut(const float* __restrict__ pl,   // (2,2,7) ascending
                             const float* __restrict__ ph,   // (2,2,6) ascending
                             float* __restrict__ lut_g) {
  int qi = threadIdx.x;
  if (qi >= LUT_N) return;
  float w = (float)(qi - 128) * WSCALE;
#pragma unroll
  for (int j = 0; j < NCOMBO; ++j) {
    const float* a = pl + j * 7;
    const float* b = ph + j * 6;
    float lo = a[6];
#pragma unroll
    for (int d = 5; d >= 0; --d) lo = __fmaf_rn(lo, w, a[d]);
    float hi = b[5];
#pragma unroll
    for (int d = 4; d >= 0; --d) hi = __fmaf_rn(hi, w, b[d]);
    lut_g[qi * 8 + j * 2 + 0] = lo;
    lut_g[qi * 8 + j * 2 + 1] = hi - lo;
  }
}

// ---------------- kernel B: quantize + pad + batch-interleave ----------------
// qpack[c][tau][b] (u16 level in [0,256]); tau in [0,430)
__global__ void mc_build_qpack(const float* __restrict__ x,
                               unsigned short* __restrict__ qpack) {
  int idx = blockIdx.x * blockDim.x + threadIdx.x; // over C*TPAD
  if (idx >= Cn * TPAD) return;
  int c = idx / TPAD, tau = idx % TPAD;
  int ts = tau - 15; // left pad = 15
  ushort4 v;
  unsigned short qv[4];
#pragma unroll
  for (int b = 0; b < Bn; ++b) {
    float xv = (ts >= 0 && ts < Tn) ? x[((size_t)b * Cn + c) * Tn + ts] : 0.0f;
    float q = rintf(xv * 128.0f);                  // 1/DAC_SCALE
    q = fminf(fmaxf(q, -128.0f), 128.0f);
    qv[b] = (unsigned short)(int)(q + 128.0f);     // level index
  }
  v.x = qv[0]; v.y = qv[1]; v.z = qv[2]; v.w = qv[3];
  *(ushort4*)&qpack[(size_t)idx * 4] = v;
}

// ---------------- main kernel ----------------
__global__ void __launch_bounds__(TILE_T)
mc_main(const unsigned short* __restrict__ qpack,
        const float* __restrict__ lut_g,
        const float* __restrict__ r,      // (2,2,C,K)
        const float* __restrict__ bias,   // (C,)
        float* __restrict__ out) {        // (B,C,T)
  __shared__ __align__(16) float lut_lds[LUT_N * 8];   // 8224 B
  __shared__ __align__(8)  unsigned short q_lds[QROW * 4]; // 2304 B
  __shared__ __align__(16) float rr_lds[NCOMBO * Kn];  // 496 B

  const int tid = threadIdx.x;
  const int c   = blockIdx.y;
  const int t0  = blockIdx.x * TILE_T;
  const int t   = t0 + tid;

#if HAVE_TDM
  if (tid == 0) {
    // LUT: 1D, 257*8 f32
    tdm_load((unsigned)(uintptr_t)(void*)lut_lds, lut_g,
             2u, LUT_N * 8u, LUT_N * 8u, 0u, 0ull);
    // q window: 1D run of (tau,b)-interleaved u16; OOB (last tile) reads zero
    tdm_load((unsigned)(uintptr_t)(void*)q_lds,
             qpack + (size_t)(c * TPAD + t0) * Bn,
             1u, (unsigned)((TPAD - t0) * Bn), QROW * Bn, 0u, 0ull);
    // rr slice: 4 rows of K f32, row stride C*K
    tdm_load((unsigned)(uintptr_t)(void*)rr_lds, r + (size_t)c * Kn,
             2u, (unsigned)Kn, (unsigned)Kn, NCOMBO,
             (unsigned long long)(Cn * Kn));
    __builtin_amdgcn_s_wait_tensorcnt(0);
  }
  __syncthreads();
#else
  for (int i = tid; i < LUT_N * 8; i += TILE_T) lut_lds[i] = lut_g[i];
  {
    const size_t gbase = (size_t)(c * TPAD + t0) * Bn;
    for (int i = tid; i < QROW * Bn; i += TILE_T)
      q_lds[i] = ((t0 * Bn + i) < TPAD * Bn) ? qpack[gbase + i] : (unsigned short)0;
  }
  for (int i = tid; i < NCOMBO * Kn; i += TILE_T) {
    int j = i / Kn, k = i % Kn;
    rr_lds[i] = r[((size_t)j * Cn + c) * Kn + k];
  }
  __syncthreads();
#endif

  if (t < Tn) {
    float acc[NCOMBO][Bn];
#pragma unroll
    for (int j = 0; j < NCOMBO; ++j)
#pragma unroll
      for (int b = 0; b < Bn; ++b) acc[j][b] = 0.0f;

    const unsigned nbase = ((unsigned)t * Cn + (unsigned)c) * Kn;

    for (int k = 0; k < Kn; ++k) {
      const float rr0 = rr_lds[0 * Kn + k];
      const float rr1 = rr_lds[1 * Kn + k];
      const float rr2 = rr_lds[2 * Kn + k];
      const float rr3 = rr_lds[3 * Kn + k];

      // 4 normals per (t,c,k), one per combo, shared across batch (as in ref).
      const unsigned ctr = (nbase + (unsigned)k) * 4u;
      const unsigned h0 = hash32(ctr + 0x9E3779B9u);
      const unsigned h1 = hash32(ctr + 0x85EBCA6Bu);
      const unsigned h2 = hash32(ctr + 0xC2B2AE35u);
      const unsigned h3 = hash32(ctr + 0x27D4EB2Fu);
      const float ra = FAST_SQRT(-2.0f * __logf(u01(h0)));
      const float rb = FAST_SQRT(-2.0f * __logf(u01(h2)));
      float sa, ca, sb, cb;
      __sincosf(TWO_PI * u01(h1), &sa, &ca);
      __sincosf(TWO_PI * u01(h3), &sb, &cb);
      const float n0 = ra * ca, n1 = ra * sa, n2 = rb * cb, n3 = rb * sb;

      // 4 batch levels in one b64 LDS load
      const uint2 qq = *(const uint2*)&q_lds[(tid + k) * 4];
      int qis[4];
      qis[0] = (int)(qq.x & 0xffffu);
      qis[1] = (int)(qq.x >> 16);
      qis[2] = (int)(qq.y & 0xffffu);
      qis[3] = (int)(qq.y >> 16);

#pragma unroll
      for (int b = 0; b < Bn; ++b) {
        const int qi = qis[b];
        const float aw   = fabsf((float)(qi - 128)) * WSCALE;
        const float afac = J_COEF * FAST_RCP(aw + EPSC) + S_COEF;
        const float4* L  = (const float4*)&lut_lds[qi * 8];
        const float4 e0 = L[0];
        const float4 e1 = L[1];
        { // combo 0: i=0,s=0
          float raw = __fmaf_rn(e0.y, rr0, e0.x);
          float sg  = FAST_SQRT(fabsf(raw) * afac);
          acc[0][b] = __fmaf_rn(n0, sg, acc[0][b] + raw);
        }
        { // combo 1: i=0,s=1
          float raw = __fmaf_rn(e0.w, rr1, e0.z);
          float sg  = FAST_SQRT(fabsf(raw) * afac);
          acc[1][b] = __fmaf_rn(n1, sg, acc[1][b] + raw);
        }
        { // combo 2: i=1,s=0
          float raw = __fmaf_rn(e1.y, rr2, e1.x);
          float sg  = FAST_SQRT(fabsf(raw) * afac);
          acc[2][b] = __fmaf_rn(n2, sg, acc[2][b] + raw);
        }
        { // combo 3: i=1,s=1
          float raw = __fmaf_rn(e1.w, rr3, e1.z);
          float sg  = FAST_SQRT(fabsf(raw) * afac);
          acc[3][b] = __fmaf_rn(n3, sg, acc[3][b] + raw);
        }
      }
    }

    const float bc = bias[c];
#pragma unroll
    for (int b = 0; b < Bn; ++b) {
      const float p0 = (acc[0][b] - acc[1][b]) * CUR_SCALEF; // bit 1
      const float p1 = (acc[2][b] - acc[3][b]) * CUR_SCALEF; // bit 0
      const float o  = adc_fq(p0) * 2.0f + adc_fq(p1);
      out[((size_t)b * Cn + c) * Tn + t] = o + bc;           // OUTPUT_FACTOR = 1
    }
  }
}

// ---------------- launcher ----------------
extern "C" void kernel_launch(void* const* d_in, const int* in_sizes, int n_in,
                              void* d_out, int out_size, void* d_ws, size_t ws_size,
                              hipStream_t stream) {
  (void)in_sizes; (void)n_in; (void)out_size; (void)ws_size;
  const float* x    = (const float*)d_in[0];  // (B,C,T)
  const float* pl   = (const float*)d_in[1];  // (2,2,7)
  const float* ph   = (const float*)d_in[2];  // (2,2,6)
  const float* r    = (const float*)d_in[3];  // (2,2,C,K)
  const float* bias = (const float*)d_in[4];  // (C,)
  float* out = (float*)d_out;

  float* lut_g = (float*)d_ws;                                 // 8224 B
  unsigned short* qpack = (unsigned short*)((char*)d_ws + 8704); // 1.76 MB

  mc_build_lut<<<1, 288, 0, stream>>>(pl, ph, lut_g);
  mc_build_qpack<<<(Cn * TPAD + 255) / 256, 256, 0, stream>>>(x, qpack);

  dim3 grid((Tn + TILE_T - 1) / TILE_T, Cn);
  mc_main<<<grid, TILE_T, 0, stream>>>(qpack, lut_g, r, bias, out);
}